// fNRI_27058293965068
// MI455X (gfx1250) — compile-verified
//
#include <hip/hip_runtime.h>

#define NB   8
#define NN   1000
#define KREL 3

typedef __attribute__((ext_vector_type(2))) float v2f;
typedef __attribute__((ext_vector_type(4))) float v4f;
typedef __attribute__((ext_vector_type(8))) float v8f;

// workspace float layout
#define WS_SX    0     // 8
#define WS_XS    16    // 1024 floats, zero padded past NN
#define WS_CONST 1056  // 32 floats
#define WS_E     1088  // 4096 floats: E[j*4+k] = exp(xs[j]*eB[k]), k<3; pad 0
// consts: [0..2] eA  [3..5] eB  [6..8] eC
//         [9..14] mE[k*2+m]  [15..20] mO  [21..26] nn  [27..29] wv  [30] bv

__global__ __launch_bounds__(256) void nri_setup(
    const float* __restrict__ x,
    const float* W1, const float* B1, const float* W2, const float* B2,
    const float* Wf1e, const float* bf1e, const float* Wf2e, const float* bf2e,
    const float* Wfk, const float* bfk, const float* Wf1v, const float* bf1v,
    const float* Wf2v, const float* bf2v,
    float* __restrict__ ws)
{
  __shared__ float Sx[NB];
  const int t = threadIdx.x;

  // xs[i] = sum_b x[b,i]  (zero padded to 1024)
  for (int i = t; i < 1024; i += 256) {
    float s = 0.f;
    if (i < NN) {
      for (int b = 0; b < NB; ++b) s += x[b * NN + i];
    }
    ws[WS_XS + i] = s;
  }
  // Sx[b] = sum_j x[b,j]
  if (t < NB) {
    float s = 0.f;
    const float* xr = x + t * NN;
    for (int j = 0; j < NN; ++j) s += xr[j];
    Sx[t] = s;
    ws[WS_SX + t] = s;
  }
  __syncthreads();

  if (t == 0) {
    const float Nf = (float)NN;
    float cE[2], cO[2], dE[2], dO[2], hA[2];
    for (int f = 0; f < 2; ++f) {
      float ce = 0, co = 0, de = 0, dd = 0;
      for (int e = 0; e < 4; ++e) {
        ce += W1[e] * Wf1e[(2 * e) * 2 + f];
        co += W1[e] * Wf1e[(2 * e + 1) * 2 + f];
        de += B1[e] * Wf1e[(2 * e) * 2 + f];
        dd += B1[e] * Wf1e[(2 * e + 1) * 2 + f];
      }
      cE[f] = ce; cO[f] = co; dE[f] = de; dO[f] = dd;
      hA[f] = Nf * ce;
    }
    float h2A[3], G[3];
    for (int v = 0; v < 3; ++v) {
      float a = 0;
      for (int f = 0; f < 2; ++f) a += hA[f] * Wf1v[f * 3 + v];
      h2A[v] = a;
      float g = 0;
      for (int b = 0; b < NB; ++b) {
        float hv = 0;
        for (int f = 0; f < 2; ++f) {
          float hBbf = Sx[b] * cO[f] + Nf * (dE[f] + dO[f] + bf1e[f]);
          hv += hBbf * Wf1v[f * 3 + v];
        }
        g += hv + bf1v[v];
      }
      G[v] = g;
    }
    float* C = ws + WS_CONST;
    for (int k = 0; k < KREL; ++k) {
      float ea = 0, eb = 0, ec = 0;
      for (int v = 0; v < 3; ++v) {
        ea += h2A[v] * Wf2e[(2 * v) * 3 + k];
        eb += h2A[v] * Wf2e[(2 * v + 1) * 3 + k];
        ec += G[v] * (Wf2e[(2 * v) * 3 + k] + Wf2e[(2 * v + 1) * 3 + k]);
      }
      C[0 + k] = ea; C[3 + k] = eb; C[6 + k] = ec + bf2e[k];
      for (int m = 0; m < 2; ++m) {
        float me = 0, mo = 0, nn = 0;
        for (int e = 0; e < 4; ++e) {
          me += W2[e] * Wfk[k * 16 + (2 * e) * 2 + m];
          mo += W2[e] * Wfk[k * 16 + (2 * e + 1) * 2 + m];
          nn += B2[e] * (Wfk[k * 16 + (2 * e) * 2 + m] +
                         Wfk[k * 16 + (2 * e + 1) * 2 + m]);
        }
        C[9 + k * 2 + m]  = me;
        C[15 + k * 2 + m] = mo;
        C[21 + k * 2 + m] = nn + bfk[k * 2 + m];
      }
    }
    C[27] = Wf2v[0]; C[28] = Wf2v[1]; C[29] = Wf2v[2]; C[30] = bf2v[0];
  }
  __syncthreads();

  // E[j][k] = exp(xs[j] * eB[k])  (per-j softmax numerator factors)
  {
    const float eB0 = ws[WS_CONST + 3];
    const float eB1 = ws[WS_CONST + 4];
    const float eB2 = ws[WS_CONST + 5];
    for (int j = t; j < 1024; j += 256) {
      float xsj = ws[WS_XS + j];
      ws[WS_E + j * 4 + 0] = __expf(xsj * eB0);
      ws[WS_E + j * 4 + 1] = __expf(xsj * eB1);
      ws[WS_E + j * 4 + 2] = __expf(xsj * eB2);
      ws[WS_E + j * 4 + 3] = 0.f;
    }
  }
}

// one block per 16-row i-tile; 8 wave32 per block stride the j range.
__global__ __launch_bounds__(256) void nri_main(
    const float* __restrict__ x, const float* __restrict__ ws,
    float* __restrict__ out)
{
  __shared__ float Elds[1024 * 4];       // 16 KB: E[j*4+k]
  __shared__ float red[8 * 3 * 8 * 32];  // 24 KB: per-wave C dumps
  __shared__ float fin[3 * 8 * 32];      //  3 KB: reduced C

  const int tid  = threadIdx.x;
  const int wave = tid >> 5;
  const int lane = tid & 31;
  const int i0   = blockIdx.x * 16;

  for (int e = tid; e < 1024 * 4; e += 256) Elds[e] = ws[WS_E + e];
  __syncthreads();

  const float* C = ws + WS_CONST;
  const float eA0 = C[0], eA1 = C[1], eA2 = C[2];
  const float eC0 = C[6], eC1 = C[7], eC2 = C[8];

  // A-operand (16x4 f32): lanes 0-15 -> rows M=lane, j-cols {0,1};
  // lanes 16-31 -> rows M=lane-16, j-cols {2,3}. B mirrored: VGPR0 -> j-row
  // {0|2}, VGPR1 -> j-row {1|3}, col N = lane&15.
  const int m    = lane & 15;
  const int hi   = lane >> 4;
  const int ncol = m;
  const float xsi = ws[WS_XS + i0 + m];   // padded region is zeroed

  // row factors R[k] = exp(xs_i*eA[k] + eC[k]); logits are O(0.1), no max-sub
  const float R0 = __expf(xsi * eA0 + eC0);
  const float R1 = __expf(xsi * eA1 + eC1);
  const float R2 = __expf(xsi * eA2 + eC2);

  // B operand: Xext cols 0..7 = x[b][j], col 8 = 1.0 (row-sum), 9..15 = 0
  const float* xrow = x + (ncol & 7) * NN;     // clamped, always valid
  const float colfill = (ncol == 8) ? 1.0f : 0.0f;
  const bool  use_x   = (ncol < 8);

  v8f c0 = {}; v8f c1 = {}; v8f c2 = {};

#pragma unroll 2
  for (int step = wave; step * 4 < NN; step += 8) {
    const int ja = step * 4 + hi * 2;    // this lane's first j (A col / B row)

    const v4f Ea = *(const v4f*)&Elds[ja * 4];
    const v4f Eb = *(const v4f*)&Elds[(ja + 1) * 4];
    const v2f xp = *(const v2f*)(xrow + ja);   // 8B-aligned (ja even)

    v2f a0v, a1v, a2v;
    {
      float e0 = R0 * Ea.x, e1 = R1 * Ea.y, e2 = R2 * Ea.z;
      float inv = __builtin_amdgcn_rcpf(e0 + e1 + e2);
      a0v.x = e0 * inv; a1v.x = e1 * inv; a2v.x = e2 * inv;
    }
    {
      float e0 = R0 * Eb.x, e1 = R1 * Eb.y, e2 = R2 * Eb.z;
      float inv = __builtin_amdgcn_rcpf(e0 + e1 + e2);
      a0v.y = e0 * inv; a1v.y = e1 * inv; a2v.y = e2 * inv;
    }

    v2f bm;
    bm.x = use_x ? xp.x : colfill;   // value-select, no exec branch
    bm.y = use_x ? xp.y : colfill;

    c0 = __builtin_amdgcn_wmma_f32_16x16x4_f32(false, a0v, false, bm,
                                               (short)0, c0, false, false);
    c1 = __builtin_amdgcn_wmma_f32_16x16x4_f32(false, a1v, false, bm,
                                               (short)0, c1, false, false);
    c2 = __builtin_amdgcn_wmma_f32_16x16x4_f32(false, a2v, false, bm,
                                               (short)0, c2, false, false);
  }

  // dump per-wave accumulators, sum across the 8 waves in fixed order (deterministic)
  for (int r = 0; r < 8; ++r) {
    red[wave * 768 + (0 * 8 + r) * 32 + lane] = c0[r];
    red[wave * 768 + (1 * 8 + r) * 32 + lane] = c1[r];
    red[wave * 768 + (2 * 8 + r) * 32 + lane] = c2[r];
  }
  __syncthreads();
  for (int e = tid; e < 768; e += 256) {
    float s = 0.f;
    for (int w = 0; w < 8; ++w) s += red[w * 768 + e];
    fin[e] = s;
  }
  __syncthreads();

  // finalize: C/D layout -> (M=mm, N=n): r = mm&7, lane = n + (mm>=8 ? 16:0)
  if (tid < 128) {
    const int b  = tid & 7;
    const int mm = tid >> 3;
    const int i  = i0 + mm;
    if (i < NN) {
      const int lg = (mm >= 8) ? 16 : 0;
      const int r  = mm & 7;
      const float xbi = x[b * NN + i];
      float ve0 = 0.f, ve1 = 0.f;
      for (int k = 0; k < KREL; ++k) {
        float sP = fin[(k * 8 + r) * 32 + lg + 8];
        float V  = fin[(k * 8 + r) * 32 + lg + b];
        float mE0 = C[9 + k * 2 + 0],  mE1 = C[9 + k * 2 + 1];
        float mO0 = C[15 + k * 2 + 0], mO1 = C[15 + k * 2 + 1];
        float nn0 = C[21 + k * 2 + 0], nn1 = C[21 + k * 2 + 1];
        ve0 += sP * (xbi * mE0 + nn0) + V * mO0;
        ve1 += sP * (xbi * mE1 + nn1) + V * mO1;
      }
      out[b * NN + i] = xbi + ve0 * C[27] + ve1 * C[28] + xbi * C[29] + C[30];
    }
  }
}

extern "C" void kernel_launch(void* const* d_in, const int* in_sizes, int n_in,
                              void* d_out, int out_size, void* d_ws, size_t ws_size,
                              hipStream_t stream) {
  (void)in_sizes; (void)n_in; (void)out_size; (void)ws_size;
  const float* x    = (const float*)d_in[0];
  const float* W1   = (const float*)d_in[1];
  const float* B1   = (const float*)d_in[2];
  const float* W2   = (const float*)d_in[3];
  const float* B2   = (const float*)d_in[4];
  const float* Wf1e = (const float*)d_in[5];
  const float* bf1e = (const float*)d_in[6];
  const float* Wf2e = (const float*)d_in[7];
  const float* bf2e = (const float*)d_in[8];
  const float* Wfk  = (const float*)d_in[9];
  const float* bfk  = (const float*)d_in[10];
  const float* Wf1v = (const float*)d_in[11];
  const float* bf1v = (const float*)d_in[12];
  const float* Wf2v = (const float*)d_in[13];
  const float* bf2v = (const float*)d_in[14];
  float* ws  = (float*)d_ws;
  float* out = (float*)d_out;

  nri_setup<<<1, 256, 0, stream>>>(x, W1, B1, W2, B2, Wf1e, bf1e, Wf2e, bf2e,
                                   Wfk, bfk, Wf1v, bf1v, Wf2v, bf2v, ws);
  const int tiles = (NN + 15) / 16;  // 63
  nri_main<<<tiles, 256, 0, stream>>>(x, ws, out);
}